// annotate_model_15874199126168
// MI455X (gfx1250) — compile-verified
//
#include <hip/hip_runtime.h>
#include <math.h>

#define NN 50000
#define NE 500000
#define CEILD(a,b) ((unsigned)(((long long)(a) + (b) - 1) / (b)))

typedef float v2f __attribute__((ext_vector_type(2)));
typedef float v8f __attribute__((ext_vector_type(8)));

__device__ __forceinline__ float wsum(float v) {
#pragma unroll
  for (int m = 16; m > 0; m >>= 1) v += __shfl_xor(v, m, 32);
  return v;
}

// ---------------- utility ----------------
__global__ void k_zero(float* __restrict__ p, long n) {
  long i = (long)blockIdx.x * blockDim.x + threadIdx.x;
  if (i < n) p[i] = 0.f;
}

__global__ void k_deg(const int* __restrict__ ei, float* __restrict__ deg) {
  int e = blockIdx.x * blockDim.x + threadIdx.x;
  if (e >= NE) return;
  unsafeAtomicAdd(&deg[ei[NE + e]], 1.0f);
}

// one wave per edge (DIM=128) / half-wave (DIM=64); float4 gather + fp32 atomics
template <int DIM>
__global__ void k_scatter(const int* __restrict__ ei, const float* __restrict__ h,
                          float* __restrict__ agg) {
  const int G = DIM / 4;
  long t = (long)blockIdx.x * blockDim.x + threadIdx.x;
  if (t >= (long)NE * G) return;
  int e = (int)(t / G), g = (int)(t % G);
  int src = ei[e], dst = ei[NE + e];
  const float4 v = *(const float4*)(h + (long)src * DIM + g * 4);
  float* p = agg + (long)dst * DIM + g * 4;
  unsafeAtomicAdd(p + 0, v.x);
  unsafeAtomicAdd(p + 1, v.y);
  unsafeAtomicAdd(p + 2, v.z);
  unsafeAtomicAdd(p + 3, v.w);
}

template <int DIM>
__global__ void k_divdeg(float* __restrict__ agg, const float* __restrict__ deg) {
  long i = (long)blockIdx.x * blockDim.x + threadIdx.x;
  if (i >= (long)NN * DIM) return;
  agg[i] /= fmaxf(deg[i / DIM], 1.0f);
}

// ---------------- WMMA GEMM ----------------
// out[M, dout_actual] = scale * (A1@W1^T + A2@W2^T + bias)
// W stored [DOUT, K] row-major and MUST have at least DOUT rows (zero-padded if
// the logical dout is smaller) so B fragments load unconditionally — keeps the
// K-loop free of exec-mask edits (WMMA needs EXEC all-1s anyway).
// M = 50000 = 3125 * 16 exactly, so no row guard needed.
template <int DOUT>
__global__ __launch_bounds__(128) void k_gemm(
    const float* __restrict__ A1, const float* __restrict__ W1,
    const float* __restrict__ A2, const float* __restrict__ W2,
    const float* __restrict__ bias, float* __restrict__ out,
    int K, int dout_actual, int ldc, float scale) {
  constexpr int NT = DOUT / 16;
  const int wave = threadIdx.x >> 5;
  const int lane = threadIdx.x & 31;
  const int hf = lane >> 4;     // half-wave selects K pair
  const int lr = lane & 15;
  const int mtile = blockIdx.x * 4 + wave;
  if (mtile >= NN / 16) return;  // wave-uniform exit; EXEC stays all-1s for WMMA
  const int row = mtile * 16 + lr;

  v8f acc[NT];
#pragma unroll
  for (int t = 0; t < NT; ++t) acc[t] = (v8f){0.f, 0.f, 0.f, 0.f, 0.f, 0.f, 0.f, 0.f};

  for (int pass = 0; pass < 2; ++pass) {
    const float* A = pass ? A2 : A1;
    const float* W = pass ? W2 : W1;
    if (!A) continue;
    const float* arow = A + (long)row * K + 2 * hf;
    const float* wrow = W + (long)(lr)*K + 2 * hf;  // per-lane column base
    for (int k = 0; k < K; k += 4) {
      // A 16x4 frag: lane<16 -> K={k,k+1}, lane>=16 -> K={k+2,k+3}
      v2f a = *(const v2f*)(arow + k);
#pragma unroll
      for (int t = 0; t < NT; ++t) {
        v2f b = *(const v2f*)(wrow + (long)t * 16 * K + k);
        acc[t] = __builtin_amdgcn_wmma_f32_16x16x4_f32(
            false, a, false, b, (short)0, acc[t], false, false);
      }
    }
  }

#pragma unroll
  for (int t = 0; t < NT; ++t) {
    int n = t * 16 + lr;
    if (n >= dout_actual) continue;  // epilogue-only guard (pad columns dropped)
    float bv = bias ? bias[n] : 0.f;
#pragma unroll
    for (int v = 0; v < 8; ++v) {
      int r = mtile * 16 + v + 8 * hf;
      out[(long)r * ldc + n] = scale * (acc[t][v] + bv);
    }
  }
}

// ---------------- row-wise ops (one wave per row, wave32) ----------------
__global__ void k_relu_norm64(const float* __restrict__ in, float* __restrict__ o) {
  int row = blockIdx.x * (blockDim.x >> 5) + (threadIdx.x >> 5);
  if (row >= NN) return;
  int lane = threadIdx.x & 31;
  const float* p = in + (long)row * 64;
  float a = fmaxf(p[lane], 0.f), b = fmaxf(p[lane + 32], 0.f);
  float inv = 1.f / fmaxf(sqrtf(wsum(a * a + b * b)), 1e-12f);
  float* q = o + (long)row * 64;
  q[lane] = a * inv;
  q[lane + 32] = b * inv;
}

__global__ void k_norm64(const float* __restrict__ in, float* __restrict__ o) {
  int row = blockIdx.x * (blockDim.x >> 5) + (threadIdx.x >> 5);
  if (row >= NN) return;
  int lane = threadIdx.x & 31;
  const float* p = in + (long)row * 64;
  float a = p[lane], b = p[lane + 32];
  float inv = 1.f / fmaxf(sqrtf(wsum(a * a + b * b)), 1e-12f);
  float* q = o + (long)row * 64;
  q[lane] = a * inv;
  q[lane + 32] = b * inv;
}

__global__ void k_ln_gelu128(float* __restrict__ h, const float* __restrict__ g,
                             const float* __restrict__ bb) {
  int row = blockIdx.x * (blockDim.x >> 5) + (threadIdx.x >> 5);
  if (row >= NN) return;
  int lane = threadIdx.x & 31;
  float* p = h + (long)row * 128;
  float4 v = *(float4*)(p + lane * 4);
  float mu = wsum(v.x + v.y + v.z + v.w) * (1.f / 128.f);
  float d0 = v.x - mu, d1 = v.y - mu, d2 = v.z - mu, d3 = v.w - mu;
  float var = wsum(d0 * d0 + d1 * d1 + d2 * d2 + d3 * d3) * (1.f / 128.f);
  float rs = rsqrtf(var + 1e-5f);
  float dd[4] = {d0, d1, d2, d3};
  float4 ov;
  float* op = (float*)&ov;
#pragma unroll
  for (int i = 0; i < 4; ++i) {
    float y = dd[i] * rs * g[lane * 4 + i] + bb[lane * 4 + i];
    op[i] = 0.5f * y * (1.f + erff(y * 0.70710678118654752f));  // exact GELU
  }
  *(float4*)(p + lane * 4) = ov;
}

// z = LayerNorm(2*z0)*g + b, in place, dim 64
__global__ void k_resln64(float* __restrict__ z, const float* __restrict__ g,
                          const float* __restrict__ bb) {
  int row = blockIdx.x * (blockDim.x >> 5) + (threadIdx.x >> 5);
  if (row >= NN) return;
  int lane = threadIdx.x & 31;
  float* p = z + (long)row * 64;
  float a = 2.f * p[lane], b = 2.f * p[lane + 32];
  float mu = wsum(a + b) * (1.f / 64.f);
  float da = a - mu, db = b - mu;
  float var = wsum(da * da + db * db) * (1.f / 64.f);
  float rs = rsqrtf(var + 1e-5f);
  p[lane] = da * rs * g[lane] + bb[lane];
  p[lane + 32] = db * rs * g[lane + 32] + bb[lane + 32];
}

// d = silu(t); d /= max(||d||,eps)  (in place, dim 128)
__global__ void k_silu_norm128(float* __restrict__ h) {
  int row = blockIdx.x * (blockDim.x >> 5) + (threadIdx.x >> 5);
  if (row >= NN) return;
  int lane = threadIdx.x & 31;
  float* p = h + (long)row * 128;
  float4 v = *(float4*)(p + lane * 4);
  float x0 = v.x * (1.f / (1.f + expf(-v.x)));
  float x1 = v.y * (1.f / (1.f + expf(-v.y)));
  float x2 = v.z * (1.f / (1.f + expf(-v.z)));
  float x3 = v.w * (1.f / (1.f + expf(-v.w)));
  float inv = 1.f / fmaxf(sqrtf(wsum(x0 * x0 + x1 * x1 + x2 * x2 + x3 * x3)), 1e-12f);
  float4 ov = {x0 * inv, x1 * inv, x2 * inv, x3 * inv};
  *(float4*)(p + lane * 4) = ov;
}

// ---------------- weight prep (tiny; outputs zero-padded to tile multiples) ----
// arcWn[j] = arc_W[j] / ||row j||  (padded to [64,64], rows 50..63 = 0), grid=64
__global__ void k_arc_rownorm(const float* __restrict__ w, float* __restrict__ o) {
  int j = blockIdx.x, lane = threadIdx.x;
  float a = 0.f, b = 0.f;
  if (j < 50) {
    a = w[j * 64 + lane];
    b = w[j * 64 + lane + 32];
  }
  float inv = 1.f / fmaxf(sqrtf(wsum(a * a + b * b)), 1e-12f);
  o[j * 64 + lane] = a * inv;
  o[j * 64 + lane + 32] = b * inv;
}

// labWt[n][k] = lab_Wn[k][n] / ||col n||  (lab_Wn [64,50] -> padded [64,64]), grid=64
__global__ void k_lab_colnorm_t(const float* __restrict__ w, float* __restrict__ o) {
  int n = blockIdx.x, lane = threadIdx.x;
  float a = 0.f, b = 0.f;
  if (n < 50) {
    a = w[lane * 50 + n];
    b = w[(lane + 32) * 50 + n];
  }
  float inv = 1.f / fmaxf(sqrtf(wsum(a * a + b * b)), 1e-12f);
  o[n * 64 + lane] = a * inv;
  o[n * 64 + lane + 32] = b * inv;
}

// domWt[n][k] = dom_Wn[k][n] / ||col n||  (dom_Wn [128,4] -> padded [16,128]), grid=16
__global__ void k_dom_colnorm_t(const float* __restrict__ w, float* __restrict__ o) {
  int n = blockIdx.x, lane = threadIdx.x;
  float vals[4] = {0.f, 0.f, 0.f, 0.f};
  float s = 0.f;
  if (n < 4) {
#pragma unroll
    for (int j = 0; j < 4; ++j) {
      int k = lane + 32 * j;
      vals[j] = w[k * 4 + n];
      s += vals[j] * vals[j];
    }
  }
  float inv = 1.f / fmaxf(sqrtf(wsum(s)), 1e-12f);
#pragma unroll
  for (int j = 0; j < 4; ++j) o[n * 128 + lane + 32 * j] = vals[j] * inv;
}

// ---------------- ArcFace margin (in place on cosine) ----------------
__global__ void k_arcface(float* __restrict__ c, const int* __restrict__ label) {
  long i = (long)blockIdx.x * blockDim.x + threadIdx.x;
  if (i >= (long)NN * 50) return;
  int row = (int)(i / 50), j = (int)(i % 50);
  float cs = c[i];
  float sn = sqrtf(fmaxf(0.f, fminf(1.f, 1.f - cs * cs)));
  float phi = cs * 0.9950041652780258f - sn * 0.09983341664682815f;  // cos(m), sin(m)
  phi = (cs > -0.9950041652780258f) ? phi : (cs - 0.009983341664682815f);  // TH, MM
  float o = (label[row] == j) ? phi : cs;
  c[i] = o * 64.0f;  // S
}

// ---------------- launcher ----------------
extern "C" void kernel_launch(void* const* d_in, const int* in_sizes, int n_in,
                              void* d_out, int out_size, void* d_ws, size_t ws_size,
                              hipStream_t stream) {
  const float* x      = (const float*)d_in[0];
  const int*   ei     = (const int*)d_in[1];
  const int*   label  = (const int*)d_in[2];
  const float* se_Wl  = (const float*)d_in[3];
  const float* se_Wr  = (const float*)d_in[4];
  const float* se_b   = (const float*)d_in[5];
  const float* arc_W  = (const float*)d_in[6];
  const float* e1_Wl  = (const float*)d_in[7];
  const float* e1_Wr  = (const float*)d_in[8];
  const float* e1_b   = (const float*)d_in[9];
  const float* ln1_g  = (const float*)d_in[10];
  const float* ln1_b  = (const float*)d_in[11];
  const float* e2_Wl  = (const float*)d_in[12];
  const float* e2_Wr  = (const float*)d_in[13];
  const float* e2_b   = (const float*)d_in[14];
  const float* res_g  = (const float*)d_in[15];
  const float* res_b  = (const float*)d_in[16];
  const float* dec_Wl = (const float*)d_in[17];
  const float* dec_Wr = (const float*)d_in[18];
  const float* dec_b  = (const float*)d_in[19];
  const float* dom_W  = (const float*)d_in[20];
  const float* dom_b  = (const float*)d_in[21];
  const float* dom_Wn = (const float*)d_in[22];
  const float* lab_Wn = (const float*)d_in[23];

  // workspace carve (floats): deg | agg(128) | b64 | b128 | padded small weights
  float* ws   = (float*)d_ws;
  float* deg  = ws;
  float* agg  = ws + NN;
  float* b64  = agg + (long)NN * 128;
  float* b128 = b64 + (long)NN * 64;
  float* arcWn = b128 + (long)NN * 128;  // [64,64] zero-padded
  float* labWt = arcWn + 64 * 64;        // [64,64] zero-padded
  float* domWt = labWt + 64 * 64;        // [16,128] zero-padded

  float* out    = (float*)d_out;
  float* o_feat = out;                    // [N,64]
  float* o_arc  = out + (long)NN * 64;    // [N,50]
  float* o_z    = out + (long)NN * 114;   // [N,64]
  float* o_rec  = out + (long)NN * 178;   // [N,128]
  float* o_dom  = out + (long)NN * 306;   // [N,4]
  float* o_lab  = out + (long)NN * 310;   // [N,50]

  const unsigned GEMM_GRID = CEILD(NN / 16, 4);  // 4 waves/block, 1 tile/wave

  // ---- degree + agg_x (shared by super_encoder and encoder.conv1) ----
  k_zero<<<CEILD(NN, 256), 256, 0, stream>>>(deg, NN);
  k_zero<<<CEILD((long)NN * 128, 256), 256, 0, stream>>>(agg, (long)NN * 128);
  k_deg<<<CEILD(NE, 256), 256, 0, stream>>>(ei, deg);
  k_scatter<128><<<CEILD((long)NE * 32, 256), 256, 0, stream>>>(ei, x, agg);
  k_divdeg<128><<<CEILD((long)NN * 128, 256), 256, 0, stream>>>(agg, deg);

  // ---- super encoder: feat + ArcFace ----
  k_gemm<64><<<GEMM_GRID, 128, 0, stream>>>(agg, se_Wl, x, se_Wr, se_b, o_feat, 128, 64, 64, 1.f);
  k_relu_norm64<<<CEILD(NN, 8), 256, 0, stream>>>(o_feat, b64);
  k_arc_rownorm<<<64, 32, 0, stream>>>(arc_W, arcWn);
  k_gemm<64><<<GEMM_GRID, 128, 0, stream>>>(b64, arcWn, nullptr, nullptr, nullptr, o_arc, 64, 50, 50, 1.f);
  k_arcface<<<CEILD((long)NN * 50, 256), 256, 0, stream>>>(o_arc, label);

  // ---- encoder layer 1 (reuses agg_x) ----
  k_gemm<128><<<GEMM_GRID, 128, 0, stream>>>(agg, e1_Wl, x, e1_Wr, e1_b, b128, 128, 128, 128, 1.f);
  k_ln_gelu128<<<CEILD(NN, 8), 256, 0, stream>>>(b128, ln1_g, ln1_b);

  // ---- encoder layer 2 ----
  k_zero<<<CEILD((long)NN * 128, 256), 256, 0, stream>>>(agg, (long)NN * 128);
  k_scatter<128><<<CEILD((long)NE * 32, 256), 256, 0, stream>>>(ei, b128, agg);
  k_divdeg<128><<<CEILD((long)NN * 128, 256), 256, 0, stream>>>(agg, deg);
  k_gemm<64><<<GEMM_GRID, 128, 0, stream>>>(agg, e2_Wl, b128, e2_Wr, e2_b, o_z, 128, 64, 64, 1.f);
  k_resln64<<<CEILD(NN, 8), 256, 0, stream>>>(o_z, res_g, res_b);

  // ---- decoder ----
  k_zero<<<CEILD((long)NN * 64, 256), 256, 0, stream>>>(agg, (long)NN * 64);
  k_scatter<64><<<CEILD((long)NE * 16, 256), 256, 0, stream>>>(ei, o_z, agg);
  k_divdeg<64><<<CEILD((long)NN * 64, 256), 256, 0, stream>>>(agg, deg);
  k_gemm<128><<<GEMM_GRID, 128, 0, stream>>>(agg, dec_Wl, o_z, dec_Wr, dec_b, o_rec, 64, 128, 128, 1.f);

  // ---- domain head (GRL is identity forward at lambda=0) ----
  k_gemm<128><<<GEMM_GRID, 128, 0, stream>>>(o_z, dom_W, nullptr, nullptr, dom_b, b128, 64, 128, 128, 1.f);
  k_silu_norm128<<<CEILD(NN, 8), 256, 0, stream>>>(b128);
  k_dom_colnorm_t<<<16, 32, 0, stream>>>(dom_Wn, domWt);
  k_gemm<16><<<GEMM_GRID, 128, 0, stream>>>(b128, domWt, nullptr, nullptr, nullptr, o_dom, 128, 4, 4, 10.f);

  // ---- label head ----
  k_norm64<<<CEILD(NN, 8), 256, 0, stream>>>(o_z, b64);
  k_lab_colnorm_t<<<64, 32, 0, stream>>>(lab_Wn, labWt);
  k_gemm<64><<<GEMM_GRID, 128, 0, stream>>>(b64, labWt, nullptr, nullptr, nullptr, o_lab, 64, 50, 50, 10.f);
}